// LIF_layer_25941602467942
// MI455X (gfx1250) — compile-verified
//
#include <hip/hip_runtime.h>

// LIF spike threshold: out[i] = (x[i] >= 0.5f) ? 1.0f : 0.0f
// (V_INIT * V_DECAY == 0 folds the membrane term away; pure streaming op.)
//
// Roofline: 512 MiB traffic, 0.25 FLOP/byte -> ~23 us floor at 23.3 TB/s.
// Strategy: b128 NT loads/stores, one address calc per block-chunk with
// immediate-offset accesses, uniform (SALU) fast-path predicate, 4-deep MLP,
// wave32-friendly 256-thread blocks.

typedef float v4f __attribute__((ext_vector_type(4)));

constexpr unsigned BLOCK        = 256;              // 8 wave32s per block
constexpr unsigned UNROLL       = 4;                // 4 x b128 per thread
constexpr unsigned V4_PER_BLOCK = BLOCK * UNROLL;   // 1024 float4s per block

__device__ __forceinline__ v4f spike4(v4f v) {
    v4f s;
    s.x = (v.x >= 0.5f) ? 1.0f : 0.0f;
    s.y = (v.y >= 0.5f) ? 1.0f : 0.0f;
    s.z = (v.z >= 0.5f) ? 1.0f : 0.0f;
    s.w = (v.w >= 0.5f) ? 1.0f : 0.0f;
    return s;
}

__global__ __launch_bounds__(BLOCK) void lif_spike_v4(const v4f* __restrict__ in,
                                                      v4f* __restrict__ out,
                                                      unsigned n4) {
    // 32-bit index: lets the backend use SGPR-base + VGPR-offset addressing
    // instead of per-lane 64-bit address arithmetic.
    const unsigned base = blockIdx.x * V4_PER_BLOCK + threadIdx.x;
    const v4f* __restrict__ pin  = in  + base;
    v4f* __restrict__       pout = out + base;

    // Block-uniform predicate -> SALU compare + branch, no exec-mask churn.
    if ((blockIdx.x + 1u) * V4_PER_BLOCK <= n4) {
        // Constant offsets (k*BLOCK*16 bytes = 4096/8192/12288) fold into the
        // 24-bit instruction IOFFSET: 1 address calc, 8 immediate-offset ops.
        v4f a0 = __builtin_nontemporal_load(pin + 0u * BLOCK);
        v4f a1 = __builtin_nontemporal_load(pin + 1u * BLOCK);
        v4f a2 = __builtin_nontemporal_load(pin + 2u * BLOCK);
        v4f a3 = __builtin_nontemporal_load(pin + 3u * BLOCK);

        v4f s0 = spike4(a0);
        v4f s1 = spike4(a1);
        v4f s2 = spike4(a2);
        v4f s3 = spike4(a3);

        __builtin_nontemporal_store(s0, pout + 0u * BLOCK);
        __builtin_nontemporal_store(s1, pout + 1u * BLOCK);
        __builtin_nontemporal_store(s2, pout + 2u * BLOCK);
        __builtin_nontemporal_store(s3, pout + 3u * BLOCK);
    } else {
        // Boundary block only: guarded per-vector.
        #pragma unroll
        for (unsigned k = 0; k < UNROLL; ++k) {
            const unsigned idx = base + k * BLOCK;
            if (idx < n4) {
                v4f a = __builtin_nontemporal_load(in + idx);
                __builtin_nontemporal_store(spike4(a), out + idx);
            }
        }
    }
}

__global__ __launch_bounds__(BLOCK) void lif_spike_tail(const float* __restrict__ in,
                                                        float* __restrict__ out,
                                                        int n) {
    const int i = (int)(blockIdx.x * blockDim.x + threadIdx.x);
    if (i < n) {
        out[i] = (in[i] >= 0.5f) ? 1.0f : 0.0f;
    }
}

extern "C" void kernel_launch(void* const* d_in, const int* in_sizes, int n_in,
                              void* d_out, int out_size, void* d_ws, size_t ws_size,
                              hipStream_t stream) {
    (void)n_in; (void)d_ws; (void)ws_size; (void)out_size;

    const float* x = (const float*)d_in[0];
    float*       y = (float*)d_out;

    const int      n  = in_sizes[0];          // 67,108,864 for this shape
    const unsigned n4 = (unsigned)(n / 4);    // 16,777,216 float4s

    unsigned grid = (n4 + V4_PER_BLOCK - 1) / V4_PER_BLOCK;   // 16384 blocks
    if (grid == 0) grid = 1;

    lif_spike_v4<<<grid, BLOCK, 0, stream>>>((const v4f*)x, (v4f*)y, n4);

    const int rem = n - (int)(n4 * 4u);       // 0 for this shape
    if (rem > 0) {
        lif_spike_tail<<<1, BLOCK, 0, stream>>>(x + (size_t)n4 * 4u,
                                                y + (size_t)n4 * 4u, rem);
    }
}